// OTDistillationLoss_10290741641976
// MI455X (gfx1250) — compile-verified
//
#include <hip/hip_runtime.h>
#include <math.h>

typedef float v2f __attribute__((ext_vector_type(2)));
typedef float v8f __attribute__((ext_vector_type(8)));

#define NROWS 1024   // student rows per batch
#define MCOLS 768    // teacher rows per batch
#define DDIM  1024   // embedding dim
#define EPS_INV 10.0f           // 1/EPS, EPS = 0.1
#define SINK_ITERS 30
#define LAMBDA_OT 1.0f
#define LOG_A (-6.93147180559945f)   // -ln(1024)
#define LOG_B (-6.64378973314767f)   // -ln(768)

// ---------------- init: zero out, log_f, log_g ----------------
__global__ void k_init(float* out, float* logf, float* logg, int nf, int ng) {
    int idx = blockIdx.x * blockDim.x + threadIdx.x;
    if (idx == 0) out[0] = 0.0f;
    if (idx < nf) logf[idx] = 0.0f;
    if (idx < ng) logg[idx] = 0.0f;
}

// ---------------- row sums of squares (wave per row) ----------------
__global__ void __launch_bounds__(256) k_sumsq(const float* __restrict__ X,
                                               const float* __restrict__ Y,
                                               float* __restrict__ xsq,
                                               float* __restrict__ ysq, int B) {
    int w    = (blockIdx.x * blockDim.x + threadIdx.x) >> 5;
    int lane = threadIdx.x & 31;
    int nx = B * NROWS;
    int total = nx + B * MCOLS;
    if (w >= total) return;
    const float* src;
    float* dst;
    if (w < nx) { src = X + (size_t)w * DDIM;        dst = xsq + w; }
    else        { src = Y + (size_t)(w - nx) * DDIM; dst = ysq + (w - nx); }
    float s = 0.0f;
    for (int t = lane; t < DDIM; t += 32) { float v = src[t]; s = fmaf(v, v, s); }
    for (int off = 16; off; off >>= 1) s += __shfl_xor(s, off, 32);
    if (lane == 0) *dst = s;
}

// ---------------- cost matrix via fp32 WMMA ----------------
// Each wave computes a 64x32 tile of C_b = max(xsq_i + ysq_j - 2*X@Y^T, 0)
__global__ void __launch_bounds__(256) k_cost_wmma(const float* __restrict__ X,
                                                   const float* __restrict__ Y,
                                                   const float* __restrict__ xsq,
                                                   const float* __restrict__ ysq,
                                                   float* __restrict__ C) {
    const int lane = threadIdx.x & 31;
    const int lo = lane & 15;
    const int hi = lane >> 4;
    const int wavesPerBatch = (NROWS / 64) * (MCOLS / 32);   // 384
    int w = (blockIdx.x * blockDim.x + threadIdx.x) >> 5;
    int b = w / wavesPerBatch;
    int t = w % wavesPerBatch;
    int M0 = (t / (MCOLS / 32)) * 64;
    int N0 = (t % (MCOLS / 32)) * 32;

    const float* Xb = X + (size_t)b * NROWS * DDIM;
    const float* Yb = Y + (size_t)b * MCOLS * DDIM;

    v8f acc[4][2];
#pragma unroll
    for (int am = 0; am < 4; ++am)
#pragma unroll
        for (int bn = 0; bn < 2; ++bn) acc[am][bn] = (v8f)0.0f;

    for (int kk = 0; kk < DDIM; kk += 4) {
        int kofs = kk + hi * 2;   // 8B-aligned float2 per lane (ISA 16x4 f32 A layout)
        v2f a[4], bb[2];
#pragma unroll
        for (int am = 0; am < 4; ++am)
            a[am] = *(const v2f*)(Xb + (size_t)(M0 + am * 16 + lo) * DDIM + kofs);
#pragma unroll
        for (int bn = 0; bn < 2; ++bn)
            bb[bn] = *(const v2f*)(Yb + (size_t)(N0 + bn * 16 + lo) * DDIM + kofs);
#pragma unroll
        for (int am = 0; am < 4; ++am)
#pragma unroll
            for (int bn = 0; bn < 2; ++bn)
                acc[am][bn] = __builtin_amdgcn_wmma_f32_16x16x4_f32(
                    false, a[am], false, bb[bn], (short)0, acc[am][bn], false, false);
    }

    float* Cb = C + (size_t)b * NROWS * MCOLS;
    const float* xs = xsq + b * NROWS;
    const float* ys = ysq + b * MCOLS;
#pragma unroll
    for (int am = 0; am < 4; ++am) {
#pragma unroll
        for (int bn = 0; bn < 2; ++bn) {
            int j = N0 + bn * 16 + lo;
            float yv = ys[j];
#pragma unroll
            for (int r = 0; r < 8; ++r) {
                int i = M0 + am * 16 + hi * 8 + r;     // D-matrix VGPR layout
                float val = fmaxf(xs[i] + yv - 2.0f * acc[am][bn][r], 0.0f);
                Cb[(size_t)i * MCOLS + j] = val;
            }
        }
    }
}

// ---------------- Sinkhorn row update: log_f (wave per row) ----------------
__global__ void __launch_bounds__(256) k_row_update(const float* __restrict__ C,
                                                    const float* __restrict__ logg,
                                                    float* __restrict__ logf, int B) {
    int w    = (blockIdx.x * blockDim.x + threadIdx.x) >> 5;
    int lane = threadIdx.x & 31;
    if (w >= B * NROWS) return;
    int b = w / NROWS;
    int i = w % NROWS;
    const float* Crow = C + (size_t)b * NROWS * MCOLS + (size_t)i * MCOLS;
    const float* gg = logg + b * MCOLS;
    float m = -__builtin_inff(), s = 0.0f;
    for (int j = lane; j < MCOLS; j += 32) {
        float x = gg[j] - EPS_INV * Crow[j];
        float M = fmaxf(m, x);
        s = s * __expf(m - M) + __expf(x - M);
        m = M;
    }
    for (int off = 16; off; off >>= 1) {
        float mo = __shfl_xor(m, off, 32);
        float so = __shfl_xor(s, off, 32);
        float M = fmaxf(m, mo);
        s = s * __expf(m - M) + so * __expf(mo - M);
        m = M;
    }
    if (lane == 0) logf[w] = LOG_A - (m + __logf(s));
}

// ---------------- Sinkhorn col update: log_g (thread per column) ----------------
__global__ void __launch_bounds__(256) k_col_update(const float* __restrict__ C,
                                                    const float* __restrict__ logf,
                                                    float* __restrict__ logg, int B) {
    int idx = blockIdx.x * blockDim.x + threadIdx.x;
    if (idx >= B * MCOLS) return;
    int b = idx / MCOLS;
    int j = idx % MCOLS;
    const float* Cb = C + (size_t)b * NROWS * MCOLS;
    const float* ff = logf + b * NROWS;
    float m = -__builtin_inff(), s = 0.0f;
    for (int i = 0; i < NROWS; ++i) {
        float x = ff[i] - EPS_INV * Cb[(size_t)i * MCOLS + j];  // coalesced across threads
        float M = fmaxf(m, x);
        s = s * __expf(m - M) + __expf(x - M);
        m = M;
    }
    logg[idx] = LOG_B - (m + __logf(s));
}

// ---------------- loss: sum(plan * C), wave per row, fixed-order reduce ----------------
__global__ void __launch_bounds__(256) k_loss(const float* __restrict__ C,
                                              const float* __restrict__ logf,
                                              const float* __restrict__ logg,
                                              float* __restrict__ partials, int B) {
    int w    = (blockIdx.x * blockDim.x + threadIdx.x) >> 5;
    int lane = threadIdx.x & 31;
    float acc = 0.0f;
    if (w < B * NROWS) {
        int b = w / NROWS;
        int i = w % NROWS;
        const float* Crow = C + (size_t)b * NROWS * MCOLS + (size_t)i * MCOLS;
        const float* gg = logg + b * MCOLS;
        float lf = logf[w];
        for (int j = lane; j < MCOLS; j += 32) {
            float c = Crow[j];
            acc += __expf(lf - EPS_INV * c + gg[j]) * c;
        }
    }
    for (int off = 16; off; off >>= 1) acc += __shfl_xor(acc, off, 32);
    __shared__ float sm[8];
    if (lane == 0) sm[threadIdx.x >> 5] = acc;
    __syncthreads();
    if (threadIdx.x == 0) {
        float t = 0.0f;
        for (int q = 0; q < 8; ++q) t += sm[q];
        partials[blockIdx.x] = t;
    }
}

__global__ void k_final(const float* __restrict__ partials, int nparts,
                        float* __restrict__ out, float scale) {
    __shared__ float sm[256];
    float acc = 0.0f;
    for (int i = threadIdx.x; i < nparts; i += 256) acc += partials[i];
    sm[threadIdx.x] = acc;
    __syncthreads();
    for (int off = 128; off; off >>= 1) {
        if (threadIdx.x < off) sm[threadIdx.x] += sm[threadIdx.x + off];
        __syncthreads();
    }
    if (threadIdx.x == 0) out[0] = sm[0] * scale;
}

extern "C" void kernel_launch(void* const* d_in, const int* in_sizes, int n_in,
                              void* d_out, int out_size, void* d_ws, size_t ws_size,
                              hipStream_t stream) {
    const float* X = (const float*)d_in[0];   // [B,1024,1024]
    const float* Y = (const float*)d_in[1];   // [B,768,1024]
    float* out = (float*)d_out;
    int B = in_sizes[0] / (NROWS * DDIM);     // = 4

    float* ws = (float*)d_ws;
    size_t offC   = 0;
    size_t offXsq = offC   + (size_t)B * NROWS * MCOLS;
    size_t offYsq = offXsq + (size_t)B * NROWS;
    size_t offLf  = offYsq + (size_t)B * MCOLS;
    size_t offLg  = offLf  + (size_t)B * NROWS;
    size_t offPar = offLg  + (size_t)B * MCOLS;
    float* Cw   = ws + offC;
    float* xsq  = ws + offXsq;
    float* ysq  = ws + offYsq;
    float* lf   = ws + offLf;
    float* lg   = ws + offLg;
    float* part = ws + offPar;

    // 1) init (zero out, log_f, log_g)
    int initN = B * NROWS;
    k_init<<<(initN + 255) / 256, 256, 0, stream>>>(out, lf, lg, B * NROWS, B * MCOLS);

    // 2) row norms
    int rows = B * (NROWS + MCOLS);
    k_sumsq<<<(rows * 32 + 255) / 256, 256, 0, stream>>>(X, Y, xsq, ysq, B);

    // 3) cost matrix via fp32 WMMA
    int waves = B * (NROWS / 64) * (MCOLS / 32);   // 1536 waves
    k_cost_wmma<<<waves / 8, 256, 0, stream>>>(X, Y, xsq, ysq, Cw);

    // 4) Sinkhorn iterations
    for (int it = 0; it < SINK_ITERS; ++it) {
        k_row_update<<<B * NROWS / 8, 256, 0, stream>>>(Cw, lg, lf, B);
        k_col_update<<<(B * MCOLS + 255) / 256, 256, 0, stream>>>(Cw, lf, lg, B);
    }

    // 5) loss reduction (deterministic two-stage)
    int nblk = B * NROWS / 8;   // 512
    k_loss<<<nblk, 256, 0, stream>>>(Cw, lf, lg, part, B);
    k_final<<<1, 256, 0, stream>>>(part, nblk, out, LAMBDA_OT / (float)B);
}